// Aggregation_15152644621065
// MI455X (gfx1250) — compile-verified
//
#include <hip/hip_runtime.h>
#include <hip/hip_bf16.h>

// ---------------------------------------------------------------------------
// Problem constants (fixed in the reference module)
// ---------------------------------------------------------------------------
#define D      1024
#define HEADS  16
#define HD     64      // D / HEADS
#define S      64      // n_sents
#define E      16      // n_ents
#define B      32      // batch
#define SB     (S * B) // 2048 aggregated rows

typedef __attribute__((ext_vector_type(2))) float v2f;
typedef __attribute__((ext_vector_type(8))) float v8f;

// ---------------------------------------------------------------------------
// WMMA f32 16x16x4 tile helper.  One wave computes one 16x16 C tile of
// C = A @ B (NN) or C = A @ B^T (NT, B stored row-major [N,K]).
//
// f32 16x4 A-matrix VGPR layout (ISA 7.12.2): lanes 0-15 hold M=0..15,
// VGPR v holds K = 2*(lane/16) + v.  B (4x16) mirrors this with N across
// lanes.  Both A (row-major) and W (row-major [N,K]) give contiguous
// float2 loads per lane per K-step.
// ---------------------------------------------------------------------------
template <bool B_IS_NK>   // true: B stored [N,K] row-major (x @ W.T); false: B stored [K,N]
__device__ __forceinline__ v8f wmma_tile_f32(const float* __restrict__ A, int lda,
                                             const float* __restrict__ Bp, int ldb,
                                             int K) {
    const int lane = threadIdx.x & 31;
    const int half = lane >> 4;   // 0: K pair {0,1}; 1: K pair {2,3}
    const int lo   = lane & 15;   // M (for A) / N (for B)

    const float* arow = A + (size_t)lo * lda + 2 * half;
    const float* brow = B_IS_NK ? (Bp + (size_t)lo * ldb + 2 * half)
                                : (Bp + (size_t)(2 * half) * ldb + lo);
    v8f acc = {};
#pragma unroll 4
    for (int k = 0; k < K; k += 4) {
        v2f a = *(const v2f*)(arow + k);
        v2f b;
        if (B_IS_NK) {
            b = *(const v2f*)(brow + k);
        } else {
            b.x = brow[(size_t)k * ldb];
            b.y = brow[(size_t)(k + 1) * ldb];
        }
        // (neg_a, A, neg_b, B, c_mod, C, reuse_a, reuse_b)
        acc = __builtin_amdgcn_wmma_f32_16x16x4_f32(false, a, false, b,
                                                    (short)0, acc, false, false);
    }
    return acc;
}

// C/D 16x16 f32 layout: VGPR v, lanes 0-15 -> M=v, lanes 16-31 -> M=v+8; N = lane%16.
__device__ __forceinline__ void store_tile(float* __restrict__ C, int ldc, v8f acc,
                                           const float* __restrict__ colBias) {
    const int lane = threadIdx.x & 31;
    const int half = lane >> 4;
    const int col  = lane & 15;
    const float bv = colBias ? colBias[col] : 0.0f;
#pragma unroll
    for (int v = 0; v < 8; ++v)
        C[(size_t)(v + 8 * half) * ldc + col] = acc[v] + bv;
}

// ---------------------------------------------------------------------------
// Precompute kernels (all tiny)
// ---------------------------------------------------------------------------

// q_proj[i] = (query . Wq_in[i,:] + bq_in[i]) / sqrt(hd)
__global__ void k_qproj(const float* __restrict__ q, const float* __restrict__ Wq,
                        const float* __restrict__ bq, float* __restrict__ qp) {
    const int i = blockIdx.x * blockDim.x + threadIdx.x;   // 1024
    const float* w = Wq + (size_t)i * D;
    float acc = 0.f;
    for (int k = 0; k < D; ++k) acc += q[k] * w[k];
    qp[i] = (acc + bq[i]) * 0.125f;   // 1/sqrt(64)
}

// u[h,m] = sum_j qp[h*HD+j] * Wk_in[h*HD+j, m]
__global__ void k_u(const float* __restrict__ qp, const float* __restrict__ Wk_in,
                    float* __restrict__ u) {
    const int idx = blockIdx.x * 256 + threadIdx.x;        // 16384
    const int h = idx >> 10, m = idx & (D - 1);
    float acc = 0.f;
    for (int j = 0; j < HD; ++j)
        acc += qp[h * HD + j] * Wk_in[(size_t)(h * HD + j) * D + m];
    u[(size_t)h * D + m] = acc;
}

// Wscore[h,n] = sum_m u[h,m] * Wk_lin[m,n]
__global__ void k_wscore(const float* __restrict__ u, const float* __restrict__ Wk_lin,
                         float* __restrict__ Wscore) {
    const int idx = blockIdx.x * 256 + threadIdx.x;        // 16384
    const int h = idx >> 10, n = idx & (D - 1);
    float acc = 0.f;
    for (int m = 0; m < D; ++m)
        acc += u[(size_t)h * D + m] * Wk_lin[(size_t)m * D + n];
    Wscore[idx] = acc;
}

// score_bias[h] = sum_j qp[h*HD+j] * (Wk_in[h*HD+j,:] . bk_lin + bk_in[h*HD+j])
__global__ void k_sbias(const float* __restrict__ qp, const float* __restrict__ Wk_in,
                        const float* __restrict__ bk_lin, const float* __restrict__ bk_in,
                        float* __restrict__ sbias) {
    __shared__ float tmp[D];
    const int j = threadIdx.x;                             // 1024 threads
    float acc = 0.f;
    for (int m = 0; m < D; ++m) acc += Wk_in[(size_t)j * D + m] * bk_lin[m];
    tmp[j] = qp[j] * (acc + bk_in[j]);
    __syncthreads();
    if (j < HEADS) {
        float s = 0.f;
        for (int x = 0; x < HD; ++x) s += tmp[j * HD + x];
        sbias[j] = s;
    }
}

// bv_f[i] = Wv_in[i,:] . bv_lin + bv_in[i]
__global__ void k_bvf(const float* __restrict__ Wv_in, const float* __restrict__ bv_lin,
                      const float* __restrict__ bv_in, float* __restrict__ bvf) {
    const int i = blockIdx.x * blockDim.x + threadIdx.x;   // 1024
    float acc = 0.f;
    for (int m = 0; m < D; ++m) acc += Wv_in[(size_t)i * D + m] * bv_lin[m];
    bvf[i] = acc + bv_in[i];
}

// Wv_f = Wv_in @ Wv_lin  (NN GEMM, 1024^3, WMMA)
__global__ void k_fusev(const float* __restrict__ Wv_in, const float* __restrict__ Wv_lin,
                        float* __restrict__ Wvf) {
    const int wave = threadIdx.x >> 5;                     // 128 threads = 4 waves
    const int tile = blockIdx.x * 4 + wave;                // 4096 tiles (64 x 64)
    const int mt = tile >> 6, nt = tile & 63;
    v8f acc = wmma_tile_f32<false>(Wv_in + (size_t)mt * 16 * D, D,
                                   Wv_lin + nt * 16, D, D);
    store_tile(Wvf + (size_t)mt * 16 * D + nt * 16, D, acc, nullptr);
}

// ---------------------------------------------------------------------------
// Fused scores + softmax + entity aggregation.  One block per (s,b).
// agg[(s*B+b), h, :] = sum_e attn[h,e] * entities[s*E+e, b, :]
// ---------------------------------------------------------------------------
__global__ void k_attn(const float* __restrict__ ent, const unsigned char* __restrict__ pmask,
                       const float* __restrict__ Wscore, const float* __restrict__ sbias,
                       float* __restrict__ agg) {
    const int sb = blockIdx.x;            // 2048
    const int s = sb >> 5, b = sb & (B - 1);
    const int t = threadIdx.x;            // 256
    __shared__ float sc[E][HEADS];
    __shared__ float attn[E][HEADS + 1];

    {   // scores[e,h] = E_row . Wscore[h] + sbias[h]   (mask -> -inf)
        const int e = t >> 4, h = t & 15;
        const float* erow = ent + ((size_t)(s * E + e) * B + b) * D;
        const float* wrow = Wscore + (size_t)h * D;
        float acc = 0.f;
        for (int k = 0; k < D; k += 4)
            acc += erow[k] * wrow[k] + erow[k + 1] * wrow[k + 1]
                 + erow[k + 2] * wrow[k + 2] + erow[k + 3] * wrow[k + 3];
        acc += sbias[h];
        if (pmask[(size_t)(s * E + e) * B + b]) acc = -3.0e38f;
        sc[e][h] = acc;
    }
    __syncthreads();
    {   // softmax over e (redundant per-thread, E=16 is tiny)
        const int e = t >> 4, h = t & 15;
        float m = -3.4e38f;
        for (int e2 = 0; e2 < E; ++e2) m = fmaxf(m, sc[e2][h]);
        float den = 0.f;
        for (int e2 = 0; e2 < E; ++e2) den += __expf(sc[e2][h] - m);
        attn[e][h] = (m < -2.9e38f) ? 0.f : __expf(sc[e][h] - m) / den;
    }
    __syncthreads();
    {   // aggregate: 16 threads per head, 16-wide coalesced column sweeps
        const int h = t >> 4, c = t & 15;
        float* arow = agg + ((size_t)sb * HEADS + h) * D;
        const float* ebase = ent + ((size_t)(s * E) * B + b) * D;
        for (int i = 0; i < HD; ++i) {
            const int n = i * 16 + c;
            float acc = 0.f;
#pragma unroll
            for (int e = 0; e < E; ++e)
                acc += attn[e][h] * ebase[(size_t)e * B * D + n];
            arow[n] = acc;
        }
    }
}

// ---------------------------------------------------------------------------
// ctx[m, h*HD+d'] = agg[m,h,:] . Wv_f[h*HD+d', :] + bv_f   (block-diag NT GEMM)
// ---------------------------------------------------------------------------
__global__ void k_ctx(const float* __restrict__ agg, const float* __restrict__ Wvf,
                      const float* __restrict__ bvf, float* __restrict__ ctx) {
    const int wave = threadIdx.x >> 5;
    const int tile = blockIdx.x * 4 + wave;   // 8192 tiles: 128 mt x (16 h x 4 nt)
    const int mt = tile & 127;
    const int r  = tile >> 7;                 // 0..63
    const int h = r >> 2, nt = r & 3;
    const float* A  = agg + (size_t)mt * 16 * (HEADS * D) + (size_t)h * D;
    const float* Bw = Wvf + (size_t)(h * HD + nt * 16) * D;
    v8f acc = wmma_tile_f32<true>(A, HEADS * D, Bw, D, D);
    const int ncol = h * HD + nt * 16;
    store_tile(ctx + (size_t)mt * 16 * D + ncol, D, acc, bvf + ncol);
}

// ---------------------------------------------------------------------------
// out = ctx @ Wo^T + bo   (NT GEMM, WMMA)
// ---------------------------------------------------------------------------
__global__ void k_out(const float* __restrict__ ctx, const float* __restrict__ Wo,
                      const float* __restrict__ bo, float* __restrict__ out) {
    const int wave = threadIdx.x >> 5;
    const int tile = blockIdx.x * 4 + wave;   // 8192 tiles: 128 mt x 64 nt
    const int mt = tile & 127;
    const int nt = tile >> 7;
    v8f acc = wmma_tile_f32<true>(ctx + (size_t)mt * 16 * D, D,
                                  Wo + (size_t)nt * 16 * D, D, D);
    store_tile(out + (size_t)mt * 16 * D + nt * 16, D, acc, bo + nt * 16);
}

// ---------------------------------------------------------------------------
// Host launcher
// ---------------------------------------------------------------------------
extern "C" void kernel_launch(void* const* d_in, const int* in_sizes, int n_in,
                              void* d_out, int out_size, void* d_ws, size_t ws_size,
                              hipStream_t stream) {
    const float* ent    = (const float*)d_in[0];
    const unsigned char* pm = (const unsigned char*)d_in[1];
    // d_in[2] = n_sents (fixed = 64, baked into the kernels)
    const float* query  = (const float*)d_in[3];
    const float* Wk_lin = (const float*)d_in[4];
    const float* bk_lin = (const float*)d_in[5];
    const float* Wv_lin = (const float*)d_in[6];
    const float* bv_lin = (const float*)d_in[7];
    const float* Wq_in  = (const float*)d_in[8];
    const float* bq_in  = (const float*)d_in[9];
    const float* Wk_in  = (const float*)d_in[10];
    const float* bk_in  = (const float*)d_in[11];
    const float* Wv_in  = (const float*)d_in[12];
    const float* bv_in  = (const float*)d_in[13];
    const float* Wo     = (const float*)d_in[14];
    const float* bo     = (const float*)d_in[15];
    float* out = (float*)d_out;

    // Workspace layout (floats), all offsets multiples of 1024:
    float* ws     = (float*)d_ws;
    float* qp     = ws;                        //     1024
    float* u      = ws + 1024;                 //    16384
    float* Wscore = ws + 17408;                //    16384
    float* sbias  = ws + 33792;                //       16 (padded to 1024)
    float* bvf    = ws + 34816;                //     1024
    float* Wvf    = ws + 35840;                //  1048576 (4 MB)
    float* agg    = ws + 1084416;              // 33554432 (134 MB)
    float* ctx    = ws + 34638848;             //  2097152 (8 MB)
    (void)in_sizes; (void)n_in; (void)out_size; (void)ws_size;

    // --- tiny precompute chain -------------------------------------------
    k_qproj <<<4,   256, 0, stream>>>(query, Wq_in, bq_in, qp);
    k_u     <<<64,  256, 0, stream>>>(qp, Wk_in, u);
    k_wscore<<<64,  256, 0, stream>>>(u, Wk_lin, Wscore);
    k_sbias <<<1,  1024, 0, stream>>>(qp, Wk_in, bk_lin, bk_in, sbias);
    k_bvf   <<<4,   256, 0, stream>>>(Wv_in, bv_lin, bv_in, bvf);
    k_fusev <<<1024,128, 0, stream>>>(Wv_in, Wv_lin, Wvf);        // WMMA 1024^3

    // --- bandwidth-bound fused attention (scores+softmax+aggregate) ------
    k_attn  <<<SB,  256, 0, stream>>>(ent, pm, Wscore, sbias, agg);

    // --- WMMA GEMMs: per-head value projection, then out-projection ------
    k_ctx   <<<2048,128, 0, stream>>>(agg, Wvf, bvf, ctx);
    k_out   <<<2048,128, 0, stream>>>(ctx, Wo, bo, out);
}